// SHCSATritonPaddedSoftmaxV_55860344652358
// MI455X (gfx1250) — compile-verified
//
#include <hip/hip_runtime.h>
#include <cstddef>

typedef __attribute__((ext_vector_type(2))) float v2f;
typedef __attribute__((ext_vector_type(8))) float v8f;
typedef __attribute__((ext_vector_type(4))) int   v4i;

#define T_DIM 4096
#define C_DIM 1024
#define NF    1024
#define N3    3072
#define NEG_BIG (-1e30f)

#define BM 128   // block tile rows
#define BN 64    // block tile cols
#define BK 32    // K chunk
#define PA 36    // pitch (floats) of A tile [BM][PA]  (144B rows: 16B aligned)
#define PB 34    // pitch (floats) of B^T tile [BN][PB] (136B rows: 8B aligned)

#if __has_builtin(__builtin_amdgcn_global_load_async_to_lds_b128) && \
    __has_builtin(__builtin_amdgcn_s_wait_asynccnt)
#define HAS_ASYNC_LDS 1
#else
#define HAS_ASYNC_LDS 0
#endif

// 16-byte global -> LDS copy: async DMA path on gfx1250, sync fallback otherwise
__device__ __forceinline__ void copy16_to_lds(const float* g, float* l) {
#if HAS_ASYNC_LDS
  __builtin_amdgcn_global_load_async_to_lds_b128(
      (__attribute__((address_space(1))) v4i*)g,
      (__attribute__((address_space(3))) v4i*)l, 0, 0);
#else
  *(float4*)l = *(const float4*)g;
#endif
}

__device__ __forceinline__ void async_drain() {
#if HAS_ASYNC_LDS
  __builtin_amdgcn_s_wait_asynccnt(0);
#endif
}

// ---------------------------------------------------------------------------
// Kernel 1: qkv = x @ W + b ; q *= 1/sqrt(1024); split into q,k,v buffers
// ---------------------------------------------------------------------------
__global__ __launch_bounds__(256) void qkv_gemm_kernel(
    const float* __restrict__ x, const float* __restrict__ W,
    const float* __restrict__ bias,
    float* __restrict__ qb, float* __restrict__ kb2, float* __restrict__ vb)
{
  __shared__ float xs[2][BM * PA];
  __shared__ float ws[2][BN * PB];

  const int tid  = threadIdx.x;
  const int wave = tid >> 5;
  const int lane = tid & 31;
  const int lh   = lane >> 4;        // half-wave: 0 or 1
  const int l15  = lane & 15;
  const int koff = lh * 2;           // K sub-offset per WMMA f32 16x16x4 layout
  const int m0   = blockIdx.y * BM;
  const int n0   = blockIdx.x * BN;

  v8f acc[4];
#pragma unroll
  for (int i = 0; i < 4; ++i) acc[i] = (v8f){};

  auto stage = [&](int buf, int kb) {
    // x tile [BM][BK]: async 16B copies, layout preserved
#pragma unroll
    for (int i = 0; i < 4; ++i) {
      int slot = tid + i * 256;            // 0..1023, 8 float4 per row
      int r = slot >> 3, c4 = slot & 7;
      copy16_to_lds(x + (size_t)(m0 + r) * C_DIM + kb + c4 * 4,
                    &xs[buf][r * PA + c4 * 4]);
    }
    // W tile [BK][BN] transposed -> ws[n][k] (needs VGPR transpose)
#pragma unroll
    for (int i = 0; i < 2; ++i) {
      int slot = tid + i * 256;            // 0..511, 16 float4 per row
      int r = slot >> 4, c4 = slot & 15;
      float4 vv = *(const float4*)(W + (size_t)(kb + r) * N3 + n0 + c4 * 4);
      ws[buf][(c4 * 4 + 0) * PB + r] = vv.x;
      ws[buf][(c4 * 4 + 1) * PB + r] = vv.y;
      ws[buf][(c4 * 4 + 2) * PB + r] = vv.z;
      ws[buf][(c4 * 4 + 3) * PB + r] = vv.w;
    }
  };

  stage(0, 0);
  for (int kb = 0; kb < C_DIM; kb += BK) {
    const int cur = (kb >> 5) & 1;
    async_drain();
    __syncthreads();                       // cur buffer fully resident
    if (kb + BK < C_DIM) stage(cur ^ 1, kb + BK);   // overlaps compute below

    const float* arow = &xs[cur][(wave * 16 + l15) * PA];
    const float* brow = &ws[cur][l15 * PB];
#pragma unroll
    for (int kk = 0; kk < BK; kk += 4) {
      v2f a = *(const v2f*)(arow + kk + koff);
#pragma unroll
      for (int nt = 0; nt < 4; ++nt) {
        v2f b = *(const v2f*)(brow + nt * 16 * PB + kk + koff);
        acc[nt] = __builtin_amdgcn_wmma_f32_16x16x4_f32(
            false, a, false, b, (short)0, acc[nt], false, false);
      }
    }
  }

  // epilogue: C/D layout -> row = base + r + 8*lh, col = n0 + nt*16 + l15
  const int row_base = m0 + wave * 16 + lh * 8;
#pragma unroll
  for (int nt = 0; nt < 4; ++nt) {
    int col = n0 + nt * 16 + l15;
    float bv = bias[col];
#pragma unroll
    for (int r = 0; r < 8; ++r) {
      int row = row_base + r;
      float val = acc[nt][r] + bv;
      if (col < NF)            qb [(size_t)row * NF + col]             = val * 0.03125f;
      else if (col < 2 * NF)   kb2[(size_t)row * NF + (col - NF)]      = val;
      else                     vb [(size_t)row * NF + (col - 2 * NF)]  = val;
    }
  }
}

// ---------------------------------------------------------------------------
// Kernel 2: S = q @ k^T with causal + padding mask (fully-masked tiles skipped)
// ---------------------------------------------------------------------------
__global__ __launch_bounds__(256) void scores_kernel(
    const float* __restrict__ q, const float* __restrict__ k,
    float* __restrict__ S, const int* __restrict__ n_padd_p)
{
  __shared__ float qs[2][BM * PA];
  __shared__ float ks[2][BN * PA];   // k tile is already [n][c] row-major

  const int tid  = threadIdx.x;
  const int wave = tid >> 5;
  const int lane = tid & 31;
  const int lh   = lane >> 4;
  const int l15  = lane & 15;
  const int koff = lh * 2;
  const int m0   = blockIdx.y * BM;
  const int n0   = blockIdx.x * BN;
  const int npad = n_padd_p[0];

  // whole tile masked? (uniform branch)
  if ((n0 >= m0 + BM) || (m0 + BM <= npad) || (n0 + BN <= npad)) {
    float4 neg4 = {NEG_BIG, NEG_BIG, NEG_BIG, NEG_BIG};
#pragma unroll
    for (int i = 0; i < 8; ++i) {
      int slot = tid + i * 256;          // 0..2047, 16 float4 per row
      int r = slot >> 4, c4 = slot & 15;
      *(float4*)(S + (size_t)(m0 + r) * T_DIM + n0 + c4 * 4) = neg4;
    }
    return;
  }

  v8f acc[4];
#pragma unroll
  for (int i = 0; i < 4; ++i) acc[i] = (v8f){};

  auto stage = [&](int buf, int kb) {
#pragma unroll
    for (int i = 0; i < 4; ++i) {
      int slot = tid + i * 256;
      int r = slot >> 3, c4 = slot & 7;
      copy16_to_lds(q + (size_t)(m0 + r) * NF + kb + c4 * 4,
                    &qs[buf][r * PA + c4 * 4]);
    }
#pragma unroll
    for (int i = 0; i < 2; ++i) {
      int slot = tid + i * 256;          // 0..511, BN rows x 8 float4
      int r = slot >> 3, c4 = slot & 7;
      copy16_to_lds(k + (size_t)(n0 + r) * NF + kb + c4 * 4,
                    &ks[buf][r * PA + c4 * 4]);
    }
  };

  stage(0, 0);
  for (int kb = 0; kb < NF; kb += BK) {
    const int cur = (kb >> 5) & 1;
    async_drain();
    __syncthreads();
    if (kb + BK < NF) stage(cur ^ 1, kb + BK);

    const float* arow = &qs[cur][(wave * 16 + l15) * PA];
    const float* brow = &ks[cur][l15 * PA];
#pragma unroll
    for (int kk = 0; kk < BK; kk += 4) {
      v2f a = *(const v2f*)(arow + kk + koff);
#pragma unroll
      for (int nt = 0; nt < 4; ++nt) {
        v2f b = *(const v2f*)(brow + nt * 16 * PA + kk + koff);
        acc[nt] = __builtin_amdgcn_wmma_f32_16x16x4_f32(
            false, a, false, b, (short)0, acc[nt], false, false);
      }
    }
  }

  const int row_base = m0 + wave * 16 + lh * 8;
#pragma unroll
  for (int nt = 0; nt < 4; ++nt) {
    int col = n0 + nt * 16 + l15;
#pragma unroll
    for (int r = 0; r < 8; ++r) {
      int row = row_base + r;
      bool ok = (col <= row) && (col >= npad) && (row >= npad);
      S[(size_t)row * T_DIM + col] = ok ? acc[nt][r] : NEG_BIG;
    }
  }
}

// ---------------------------------------------------------------------------
// Kernel 3: row-wise softmax of S in place (one block per row)
// ---------------------------------------------------------------------------
__global__ __launch_bounds__(256) void softmax_kernel(float* __restrict__ S)
{
  __shared__ float red[256];
  const int tid = threadIdx.x;
  float* row = S + (size_t)blockIdx.x * T_DIM;

  float4 v[4];
#pragma unroll
  for (int i = 0; i < 4; ++i) v[i] = ((const float4*)row)[tid + i * 256];

  float m = NEG_BIG;
#pragma unroll
  for (int i = 0; i < 4; ++i) {
    m = fmaxf(m, fmaxf(fmaxf(v[i].x, v[i].y), fmaxf(v[i].z, v[i].w)));
  }
  red[tid] = m;
  __syncthreads();
#pragma unroll
  for (int s = 128; s > 0; s >>= 1) {
    if (tid < s) red[tid] = fmaxf(red[tid], red[tid + s]);
    __syncthreads();
  }
  m = red[0];
  __syncthreads();

  float sum = 0.0f;
#pragma unroll
  for (int i = 0; i < 4; ++i) {
    v[i].x = __expf(v[i].x - m); v[i].y = __expf(v[i].y - m);
    v[i].z = __expf(v[i].z - m); v[i].w = __expf(v[i].w - m);
    sum += v[i].x + v[i].y + v[i].z + v[i].w;
  }
  red[tid] = sum;
  __syncthreads();
#pragma unroll
  for (int s = 128; s > 0; s >>= 1) {
    if (tid < s) red[tid] += red[tid + s];
    __syncthreads();
  }
  float inv = 1.0f / red[0];

#pragma unroll
  for (int i = 0; i < 4; ++i) {
    float4 o = {v[i].x * inv, v[i].y * inv, v[i].z * inv, v[i].w * inv};
    ((float4*)row)[tid + i * 256] = o;
  }
}

// ---------------------------------------------------------------------------
// Kernel 4: y = P @ v  (K loop truncated at causal boundary), zero padded rows
// ---------------------------------------------------------------------------
__global__ __launch_bounds__(256) void pv_gemm_kernel(
    const float* __restrict__ P, const float* __restrict__ v,
    float* __restrict__ out, const int* __restrict__ n_padd_p)
{
  __shared__ float ps[2][BM * PA];
  __shared__ float vs[2][BN * PB];

  const int tid  = threadIdx.x;
  const int wave = tid >> 5;
  const int lane = tid & 31;
  const int lh   = lane >> 4;
  const int l15  = lane & 15;
  const int koff = lh * 2;
  const int m0   = blockIdx.y * BM;
  const int n0   = blockIdx.x * BN;
  const int npad = n_padd_p[0];
  const int kmax = m0 + BM;   // P[i][j]==0 for j>i -> skip upper K chunks

  v8f acc[4];
#pragma unroll
  for (int i = 0; i < 4; ++i) acc[i] = (v8f){};

  auto stage = [&](int buf, int kb) {
#pragma unroll
    for (int i = 0; i < 4; ++i) {
      int slot = tid + i * 256;
      int r = slot >> 3, c4 = slot & 7;
      copy16_to_lds(P + (size_t)(m0 + r) * T_DIM + kb + c4 * 4,
                    &ps[buf][r * PA + c4 * 4]);
    }
#pragma unroll
    for (int i = 0; i < 2; ++i) {
      int slot = tid + i * 256;
      int r = slot >> 4, c4 = slot & 15;
      float4 vv = *(const float4*)(v + (size_t)(kb + r) * NF + n0 + c4 * 4);
      vs[buf][(c4 * 4 + 0) * PB + r] = vv.x;
      vs[buf][(c4 * 4 + 1) * PB + r] = vv.y;
      vs[buf][(c4 * 4 + 2) * PB + r] = vv.z;
      vs[buf][(c4 * 4 + 3) * PB + r] = vv.w;
    }
  };

  stage(0, 0);
  for (int kb = 0; kb < kmax; kb += BK) {
    const int cur = (kb >> 5) & 1;
    async_drain();
    __syncthreads();
    if (kb + BK < kmax) stage(cur ^ 1, kb + BK);

    const float* arow = &ps[cur][(wave * 16 + l15) * PA];
    const float* brow = &vs[cur][l15 * PB];
#pragma unroll
    for (int kk = 0; kk < BK; kk += 4) {
      v2f a = *(const v2f*)(arow + kk + koff);
#pragma unroll
      for (int nt = 0; nt < 4; ++nt) {
        v2f b = *(const v2f*)(brow + nt * 16 * PB + kk + koff);
        acc[nt] = __builtin_amdgcn_wmma_f32_16x16x4_f32(
            false, a, false, b, (short)0, acc[nt], false, false);
      }
    }
  }

  const int row_base = m0 + wave * 16 + lh * 8;
#pragma unroll
  for (int nt = 0; nt < 4; ++nt) {
    int col = n0 + nt * 16 + l15;
#pragma unroll
    for (int r = 0; r < 8; ++r) {
      int row = row_base + r;
      out[(size_t)row * NF + col] = (row >= npad) ? acc[nt][r] : 0.0f;
    }
  }
}

// ---------------------------------------------------------------------------
extern "C" void kernel_launch(void* const* d_in, const int* in_sizes, int n_in,
                              void* d_out, int out_size, void* d_ws, size_t ws_size,
                              hipStream_t stream) {
  (void)in_sizes; (void)n_in; (void)out_size; (void)ws_size;
  const float* x    = (const float*)d_in[0];
  const float* W    = (const float*)d_in[1];
  const float* bias = (const float*)d_in[2];
  const int*   npad = (const int*)d_in[3];
  float*       out  = (float*)d_out;

  float* ws   = (float*)d_ws;
  float* qbuf = ws;                                   // 4096*1024
  float* kbuf = ws + (size_t)T_DIM * NF;              // 4096*1024
  float* vbuf = ws + (size_t)2 * T_DIM * NF;          // 4096*1024
  float* Sbuf = ws + (size_t)3 * T_DIM * NF;          // 4096*4096 (64MB, L2-resident)

  dim3 blk(256);
  qkv_gemm_kernel<<<dim3(N3 / BN, T_DIM / BM), blk, 0, stream>>>(x, W, bias, qbuf, kbuf, vbuf);
  scores_kernel  <<<dim3(T_DIM / BN, T_DIM / BM), blk, 0, stream>>>(qbuf, kbuf, Sbuf, npad);
  softmax_kernel <<<dim3(T_DIM), blk, 0, stream>>>(Sbuf);
  pv_gemm_kernel <<<dim3(NF / BN, T_DIM / BM), blk, 0, stream>>>(Sbuf, vbuf, out, npad);
}